// BehaviorAwareGCNLayer_80324478370228
// MI455X (gfx1250) — compile-verified
//
#include <hip/hip_runtime.h>
#include <hip/hip_bf16.h>

// BehaviorAwareGCN layer for MI455X (gfx1250, wave32).
// Stage 1/2: fp32 WMMA GEMMs (16x16x4_f32).  Stage 3: zero accumulators.
// Stage 4: wave-per-edge gather/scale/scatter-add.  Stage 5: finalize.

typedef __attribute__((ext_vector_type(2))) float v2f;
typedef __attribute__((ext_vector_type(8))) float v8f;

#define DIM 64
#define LDS_STRIDE 68   // pad to avoid LDS bank conflicts on strided A reads

__device__ __forceinline__ float sigmoidf_(float v) {
    return 1.0f / (1.0f + __expf(-v));
}

// C = X[block 32 rows] @ Wm  (64x64), optionally scaled by sigmoid(rep[row]).
// 256 threads = 8 waves; wave w -> tile (row 16*(w>>2), col 16*(w&3)).
__global__ __launch_bounds__(256) void gemm_wmma(
    const float* __restrict__ X, const float* __restrict__ Wm,
    const float* __restrict__ rep, float* __restrict__ out,
    int n_rows, int fuse_sig) {
    __shared__ float sX[32 * LDS_STRIDE];
    __shared__ float sW[64 * LDS_STRIDE];

    const int tid = threadIdx.x;
    const int block_row0 = blockIdx.x * 32;

    // Stage W (64x64) into LDS, float4, padded stride.
    for (int i = tid; i < 64 * 16; i += 256) {
        int r = i >> 4, cq = (i & 15) * 4;
        *(float4*)&sW[r * LDS_STRIDE + cq] = *(const float4*)&Wm[r * DIM + cq];
    }
    // Stage 32-row X tile into LDS.
    for (int i = tid; i < 32 * 16; i += 256) {
        int r = i >> 4, cq = (i & 15) * 4;
        if (block_row0 + r < n_rows)
            *(float4*)&sX[r * LDS_STRIDE + cq] =
                *(const float4*)&X[(size_t)(block_row0 + r) * DIM + cq];
    }
    __syncthreads();

    const int wave = tid >> 5;
    const int lane = tid & 31;
    const int half = lane >> 4;   // 0: K=k,k+1 | 1: K=k+2,k+3 (ISA 32-bit A layout)
    const int l16  = lane & 15;
    const int row_t = (wave >> 2) * 16;   // 0 or 16 within 32-row tile
    const int col0  = (wave & 3) * 16;

    v8f c = {};
    #pragma unroll
    for (int k = 0; k < DIM; k += 4) {
        v2f a, b;
        // A fragment: 16x4, lane<16 holds K=k,k+1 for M=l16; lane>=16 K=k+2,k+3.
        const float* ap = &sX[(row_t + l16) * LDS_STRIDE + k + half * 2];
        a.x = ap[0];
        a.y = ap[1];
        // B fragment: 4x16, v0=row(k+2*half), v1=row(k+2*half+1), N=col0+l16.
        const float* bp = &sW[(k + half * 2) * LDS_STRIDE + col0 + l16];
        b.x = bp[0];
        b.y = bp[LDS_STRIDE];
        c = __builtin_amdgcn_wmma_f32_16x16x4_f32(
                /*neg_a=*/false, a, /*neg_b=*/false, b,
                /*c_mod=*/(short)0, c, /*reuse_a=*/false, /*reuse_b=*/false);
    }

    // D layout: VGPR j -> M = j + 8*half, N = l16 (ISA 32-bit C/D layout).
    const int m_base = block_row0 + row_t + half * 8;
    #pragma unroll
    for (int j = 0; j < 8; ++j) {
        int m = m_base + j;
        if (m < n_rows) {
            float v = c[j];
            if (fuse_sig) v *= sigmoidf_(rep[m]);
            out[(size_t)m * DIM + col0 + l16] = v;
        }
    }
}

__global__ void zero_f32(float* __restrict__ p, size_t n) {
    size_t i = (size_t)blockIdx.x * blockDim.x + threadIdx.x;
    if (i < n) p[i] = 0.0f;
}

// One wave32 per edge; each lane owns a float2 of the 64-dim message row.
__global__ __launch_bounds__(256) void edge_scatter(
    const long long* __restrict__ ei, const float* __restrict__ sim,
    const float* __restrict__ rep, const float* __restrict__ nsig,
    const float* __restrict__ hx, float* __restrict__ acc,
    float* __restrict__ deg, int E) {
    long long gtid = (long long)blockIdx.x * blockDim.x + threadIdx.x;
    long long e = gtid >> 5;
    int lane = threadIdx.x & 31;
    if (e >= E) return;

    int r = (int)ei[e];
    int c = (int)ei[(size_t)E + e];

    float gate = sigmoidf_(rep[r] + rep[c]);
    float coef = sim[e] * gate * nsig[c];

    const float2 v = ((const float2*)(hx + (size_t)c * DIM))[lane];
    float* dst = acc + (size_t)r * DIM + 2 * lane;
    atomicAdd(dst,     coef * v.x);
    atomicAdd(dst + 1, coef * v.y);
    if (lane == 0) atomicAdd(deg + r, 1.0f);
}

// out = leaky_relu(acc/(deg+1e-6) + out)   (out holds sigmoid(rep)*x@W_self)
__global__ void finalize_k(const float* __restrict__ acc,
                           const float* __restrict__ deg,
                           float* __restrict__ out, int n_elems) {
    int i = blockIdx.x * blockDim.x + threadIdx.x;
    if (i >= n_elems) return;
    float a = acc[i] / (deg[i >> 6] + 1e-6f);
    float o = a + out[i];
    out[i] = (o > 0.0f) ? o : 0.01f * o;
}

extern "C" void kernel_launch(void* const* d_in, const int* in_sizes, int n_in,
                              void* d_out, int out_size, void* d_ws, size_t ws_size,
                              hipStream_t stream) {
    const float*     x    = (const float*)d_in[0];
    const long long* ei   = (const long long*)d_in[1];   // int64 [2, E]
    const float*     sim  = (const float*)d_in[2];
    const float*     rep  = (const float*)d_in[3];
    const float*     nsig = (const float*)d_in[4];
    const float*     Wm   = (const float*)d_in[5];
    const float*     Ws   = (const float*)d_in[6];
    float*           out  = (float*)d_out;

    const int N = in_sizes[3];        // 100000
    const int E = in_sizes[1] / 2;    // 1600000

    // Workspace: hx [N*64] | acc [N*64] | deg [N]   (~51.6 MB)
    float* hx  = (float*)d_ws;
    float* acc = hx + (size_t)N * DIM;
    float* deg = acc + (size_t)N * DIM;

    const int row_blocks = (N + 31) / 32;
    gemm_wmma<<<row_blocks, 256, 0, stream>>>(x, Wm, rep, hx,  N, 0);
    gemm_wmma<<<row_blocks, 256, 0, stream>>>(x, Ws, rep, out, N, 1);

    const size_t zn = (size_t)N * DIM + (size_t)N;   // acc + deg contiguous
    zero_f32<<<(unsigned)((zn + 255) / 256), 256, 0, stream>>>(acc, zn);

    const long long ethreads = (long long)E * 32;
    edge_scatter<<<(unsigned)((ethreads + 255) / 256), 256, 0, stream>>>(
        ei, sim, rep, nsig, hx, acc, deg, E);

    const int n_elems = N * DIM;
    finalize_k<<<(n_elems + 255) / 256, 256, 0, stream>>>(acc, deg, out, n_elems);
}